// DiffSortNet_38603166057085
// MI455X (gfx1250) — compile-verified
//
#include <hip/hip_runtime.h>
#include <cmath>

// DiffSortNet, N=256, B=512, 36 bitonic layers.
// Reference's einsums use 0/1 selection matrices -> reduce to per-layer
// 128 disjoint 2x2 column mixes of a per-batch 256x256 LDS-resident matrix.
// CDNA5 path: 320KB WGP LDS holds the whole matrix; TDM (tensor_load_to_lds /
// tensor_store_from_lds) moves the input row in and the 256KB result tile out.

#define N_ELEM 256
#define NPAIR  128

typedef __attribute__((ext_vector_type(4))) unsigned int u32x4;
typedef __attribute__((ext_vector_type(8))) unsigned int u32x8;

// ---- Tensor DMA descriptor helpers (D#, see CDNA5 ISA ch.8 / CDNA5 ISA) ----

// Group 0: [1:0]=count=1, [63:32]=lds byte addr, [120:64]=global addr, [127:126]=type=2
__device__ __forceinline__ u32x4 tdm_group0(unsigned lds_byte_addr, unsigned long long gaddr) {
  u32x4 g;
  g.x = 1u;                                               // count=1, user mode
  g.y = lds_byte_addr;                                    // lds_addr
  g.z = (unsigned)(gaddr & 0xFFFFFFFFull);                // global_addr[31:0]
  g.w = (unsigned)((gaddr >> 32) & 0x01FFFFFFull)         // global_addr[56:32]
        | 0x80000000u;                                    // type=2 ("image")
  return g;
}

// Group 1 for a 2-D tile of 4-byte elements.
// [17:16]=data_size=2 (4B); tensor_dim0/1, tile_dim0/1, tensor_dim0_stride.
__device__ __forceinline__ u32x8 tdm_group1_2d(unsigned tensor_d0, unsigned tensor_d1,
                                               unsigned tile_d0, unsigned tile_d1,
                                               unsigned stride0_elems) {
  u32x8 g = {0u, 0u, 0u, 0u, 0u, 0u, 0u, 0u};
  g.s0 = 2u << 16;                                        // data_size = 4 bytes
  g.s1 = (tensor_d0 & 0xFFFFu) << 16;                     // tensor_dim0[15:0]
  g.s2 = ((tensor_d0 >> 16) & 0xFFFFu)                    // tensor_dim0[31:16]
       | ((tensor_d1 & 0xFFFFu) << 16);                   // tensor_dim1[15:0]
  g.s3 = ((tensor_d1 >> 16) & 0xFFFFu)                    // tensor_dim1[31:16]
       | ((tile_d0 & 0xFFFFu) << 16);                     // tile_dim0
  g.s4 = (tile_d1 & 0xFFFFu);                             // tile_dim1 (tile_dim2=0)
  g.s5 = stride0_elems;                                   // tensor_dim0_stride[31:0]
  return g;                                               // dim1_stride=0 (2-D tile)
}

__device__ __forceinline__ void tdm_load_to_lds(u32x4 g0, u32x8 g1) {
  u32x4 z = {0u, 0u, 0u, 0u};                             // groups 2/3: unused dims
  asm volatile("tensor_load_to_lds %0, %1, %2, %3"
               :: "s"(g0), "s"(g1), "s"(z), "s"(z)
               : "memory");
}

__device__ __forceinline__ void tdm_store_from_lds(u32x4 g0, u32x8 g1) {
  u32x4 z = {0u, 0u, 0u, 0u};
  asm volatile("tensor_store_from_lds %0, %1, %2, %3"
               :: "s"(g0), "s"(g1), "s"(z), "s"(z)
               : "memory");
}

// ---- Prep: dense 0/1 selection/scatter matrices -> (a, b, min_dst, max_dst) table ----
__global__ void build_pairs_kernel(const float* __restrict__ sa, const float* __restrict__ sb,
                                   const float* __restrict__ cmn, const float* __restrict__ cmx,
                                   int4* __restrict__ pairs) {
  const int l = blockIdx.x;     // layer
  const int k = threadIdx.x;    // pair (128 threads)
  const float* ra = sa + (size_t)(l * NPAIR + k) * N_ELEM;
  const float* rb = sb + (size_t)(l * NPAIR + k) * N_ELEM;
  int ai = 0, bi = 0, mn = 0, mx = 0;
  for (int j = 0; j < N_ELEM; ++j) {
    if (ra[j] > 0.5f) ai = j;
    if (rb[j] > 0.5f) bi = j;
  }
  const float* cm = cmn + (size_t)l * N_ELEM * NPAIR + k;
  const float* cx = cmx + (size_t)l * N_ELEM * NPAIR + k;
  for (int j = 0; j < N_ELEM; ++j) {
    if (cm[(size_t)j * NPAIR] > 0.5f) mn = j;
    if (cx[(size_t)j * NPAIR] > 0.5f) mx = j;
  }
  pairs[l * NPAIR + k] = make_int4(ai, bi, mn, mx);
}

// ---- Main: one batch per workgroup, X (256x256 f32) resident in LDS ----
__global__ __launch_bounds__(1024, 1)
void diffsort_kernel(const float* __restrict__ vectors, const int4* __restrict__ pairs,
                     float* __restrict__ out, int L) {
  extern __shared__ float smem[];
  float* X     = smem;                        // 256*256 floats (256 KB)
  float* alpha = X + N_ELEM * N_ELEM;         // L*128 floats
  float* vec   = alpha + L * NPAIR;           // 256 floats

  const int tid = threadIdx.x;
  const int b   = blockIdx.x;

  // Kick off TDM load of this batch's input row into LDS (async, TENSORcnt).
  if (tid == 0) {
    const unsigned lds_base = (unsigned)__builtin_amdgcn_groupstaticsize();
    const unsigned vec_lds  = lds_base +
        (unsigned)((N_ELEM * N_ELEM + L * NPAIR) * sizeof(float));
    const unsigned long long g =
        (unsigned long long)(const void*)(vectors + (size_t)b * N_ELEM);
    tdm_load_to_lds(tdm_group0(vec_lds, g),
                    tdm_group1_2d(N_ELEM, 1u, N_ELEM, 1u, N_ELEM));
  }

  // Overlap: X = identity while the DMA runs.
  for (int i = tid; i < N_ELEM * N_ELEM; i += 1024)
    X[i] = ((i >> 8) == (i & (N_ELEM - 1))) ? 1.0f : 0.0f;

  if (tid == 0) __builtin_amdgcn_s_wait_tensorcnt(0);
  __syncthreads();

  // Per-(batch,pair) mixing weights: alpha = atan(10*(vb-va))/pi + 0.5
  for (int p = tid; p < L * NPAIR; p += 1024) {
    const int4 pr = pairs[p];
    const float va = vec[pr.x];
    const float vb = vec[pr.y];
    alpha[p] = atanf(10.0f * (vb - va)) * 0.318309886183790672f + 0.5f;
  }
  __syncthreads();

  // 36 layers of 128 disjoint 2x2 column rotations, applied to every row.
  // k is loop-invariant per thread; pairs within a layer are disjoint -> no races.
  const int k    = tid & (NPAIR - 1);
  const int row0 = tid >> 7;                  // 0..7, each covers 32 rows
  for (int l = 0; l < L; ++l) {
    const int4  pr = pairs[l * NPAIR + k];
    const float al = alpha[l * NPAIR + k];
    const float om = 1.0f - al;
    for (int r = row0; r < N_ELEM; r += 8) {
      float* Xr = X + (r << 8);
      const float xa = Xr[pr.x];
      const float xb = Xr[pr.y];
      Xr[pr.z] = fmaf(al, xa, om * xb);       // w_min -> min_dst
      Xr[pr.w] = fmaf(om, xa, al * xb);       // w_max -> max_dst
    }
    __syncthreads();
  }

  // TDM store of the finished 256x256 tile (contiguous 256 KB, stride==dim0).
  if (tid == 0) {
    const unsigned x_lds = (unsigned)__builtin_amdgcn_groupstaticsize();
    const unsigned long long g =
        (unsigned long long)(void*)(out + (size_t)b * N_ELEM * N_ELEM);
    tdm_store_from_lds(tdm_group0(x_lds, g),
                       tdm_group1_2d(N_ELEM, N_ELEM, N_ELEM, N_ELEM, N_ELEM));
    __builtin_amdgcn_s_wait_tensorcnt(0);
  }
}

extern "C" void kernel_launch(void* const* d_in, const int* in_sizes, int n_in,
                              void* d_out, int out_size, void* d_ws, size_t ws_size,
                              hipStream_t stream) {
  const float* vectors = (const float*)d_in[0];   // (B, 256)
  const float* sa      = (const float*)d_in[1];   // (L, 128, 256)
  const float* sb      = (const float*)d_in[2];   // (L, 128, 256)
  const float* cmn     = (const float*)d_in[3];   // (L, 256, 128)
  const float* cmx     = (const float*)d_in[4];   // (L, 256, 128)
  float* out = (float*)d_out;                     // (B, 256, 256)

  const int B = in_sizes[0] / N_ELEM;             // 512
  const int L = in_sizes[1] / (NPAIR * N_ELEM);   // 36

  int4* pairs = (int4*)d_ws;                      // L*128*16 B = 73728 B scratch

  build_pairs_kernel<<<L, NPAIR, 0, stream>>>(sa, sb, cmn, cmx, pairs);

  const size_t smem_bytes =
      (size_t)(N_ELEM * N_ELEM + L * NPAIR + N_ELEM) * sizeof(float);  // ~275 KB
  diffsort_kernel<<<B, 1024, smem_bytes, stream>>>(vectors, pairs, out, L);
}